// TPRGANet_59734405153194
// MI455X (gfx1250) — compile-verified
//
#include <hip/hip_runtime.h>
#include <math.h>

// ---------------------------------------------------------------------------
// TPRGANet forward for MI455X (gfx1250), wave32 + V_WMMA_F32_16X16X4_F32.
// B=512, T=8, N=22, C=512, F=4096, 2 layers, 3 branches (combined via alpha).
// Padding strategy: N padded 22->32. Pad entries of the Gram are never read;
// pad-K rows in the apply matmul multiply exactly-zero att columns. So all
// out-of-range rows are handled by CLAMPING the row index (no predication,
// no exec masking in the hot loops). Gram k-loop is software-pipelined.
// ---------------------------------------------------------------------------

typedef __attribute__((ext_vector_type(2))) float v2f;
typedef __attribute__((ext_vector_type(8))) float v8f;

#define NB 512
#define NT 8
#define NN 22
#define NC 512
#define NF 4096   // NT*NC

static __device__ __forceinline__ v8f wmma_f32(v2f a, v2f b, v8f c) {
  // D = A(16x4) * B(4x16) + C(16x16), all fp32
  return __builtin_amdgcn_wmma_f32_16x16x4_f32(false, a, false, b, (short)0, c,
                                               false, false);
}

static __device__ __forceinline__ v8f v8zero() {
  v8f z = {0.f, 0.f, 0.f, 0.f, 0.f, 0.f, 0.f, 0.f};
  return z;
}

// ---------------------------------------------------------------------------
// prep: adj_norm (sym-normalized adjacency, 22x22) + alpha (softmax of
// fusion_logits, 2x3). One wave.
// ---------------------------------------------------------------------------
__global__ __launch_bounds__(32) void prep_kernel(const float* __restrict__ adj,
                                                  const float* __restrict__ flog,
                                                  float* __restrict__ adjn,
                                                  float* __restrict__ alpha) {
  __shared__ float dinv[NN];
  const int t = threadIdx.x;
  if (t < NN) {
    float rs = 0.f;
    for (int j = 0; j < NN; ++j) {
      float a = fminf(fmaxf(adj[t * NN + j], 0.f), 1.f) + (t == j ? 1.f : 0.f);
      a = fmaxf(a, 1e-8f);
      rs += a;
    }
    rs = fmaxf(rs, 1e-8f);
    float d = 1.0f / sqrtf(rs);
    dinv[t] = fminf(fmaxf(d, 0.f), 100.f);
  }
  __syncthreads();
  for (int i = t; i < NN * NN; i += 32) {
    int r = i / NN, c = i % NN;
    float a = fminf(fmaxf(adj[i], 0.f), 1.f) + (r == c ? 1.f : 0.f);
    a = fmaxf(a, 1e-8f);
    adjn[i] = dinv[r] * a * dinv[c];
  }
  if (t == 0) {
    for (int l = 0; l < 2; ++l) {
      float m = fmaxf(flog[l * 3], fmaxf(flog[l * 3 + 1], flog[l * 3 + 2]));
      float e0 = expf(flog[l * 3 + 0] - m);
      float e1 = expf(flog[l * 3 + 1] - m);
      float e2 = expf(flog[l * 3 + 2] - m);
      float s = e0 + e1 + e2;
      alpha[l * 3 + 0] = e0 / s;
      alpha[l * 3 + 1] = e1 / s;
      alpha[l * 3 + 2] = e2 / s;
    }
  }
}

// ---------------------------------------------------------------------------
// gram: per batch, G(32x32 padded) = H * H^T over K=4096 (fp32 WMMA 16x16x4).
// XL=1: H read from x with [B,T,N,C] -> [B,N,T*C] transposed addressing.
// 8 waves split K (each wave's 512-wide chunk == one t slice for XL); LDS
// reduction. For the Gram, A-frag == B-frag per tile. Pad rows (>=22) load
// from a clamped row; the resulting garbage lands only in G's pad region,
// which is never read downstream. The k-loop is double-buffered: next step's
// fragments are fetched before the current step's 4 WMMAs so the matrix ops
// cover the load latency.
// ---------------------------------------------------------------------------
template <int XL>
__global__ __launch_bounds__(256) void gram_kernel(const float* __restrict__ h,
                                                   float* __restrict__ G) {
  __shared__ float red[8][4][8][32];
  const int tid = threadIdx.x;
  const int wave = tid >> 5;
  const int lane = tid & 31;
  const int b = blockIdx.x;
  const int l15 = lane & 15;
  const bool hi = lane >= 16;
  const int row0 = l15;
  const int row1 = 16 + l15;
  const int row1c = (row1 < NN) ? row1 : (NN - 1);  // clamp, no predication

  v8f a00 = v8zero(), a01 = v8zero(), a10 = v8zero(), a11 = v8zero();

  const int kbase = wave * 512;  // one 512-wide K chunk per wave
  const int koff = (hi ? 2 : 0);
  const float* p0;
  const float* p1;
  if (XL) {
    // kbase>>9 == wave, kbase&511 == 0
    p0 = h + (((size_t)b * NT + wave) * NN + row0) * NC + koff;
    p1 = h + (((size_t)b * NT + wave) * NN + row1c) * NC + koff;
  } else {
    p0 = h + ((size_t)b * NN + row0) * NF + kbase + koff;
    p1 = h + ((size_t)b * NN + row1c) * NF + kbase + koff;
  }

  // software-pipelined: prologue load, steady state prefetches ks+1, epilogue
  v2f f0 = *(const v2f*)(p0);
  v2f f1 = *(const v2f*)(p1);
  for (int ks = 0; ks < 127; ++ks) {
    const v2f n0 = *(const v2f*)(p0 + (ks + 1) * 4);
    const v2f n1 = *(const v2f*)(p1 + (ks + 1) * 4);
    a00 = wmma_f32(f0, f0, a00);
    a01 = wmma_f32(f0, f1, a01);
    a10 = wmma_f32(f1, f0, a10);
    a11 = wmma_f32(f1, f1, a11);
    f0 = n0;
    f1 = n1;
  }
  a00 = wmma_f32(f0, f0, a00);
  a01 = wmma_f32(f0, f1, a01);
  a10 = wmma_f32(f1, f0, a10);
  a11 = wmma_f32(f1, f1, a11);

#pragma unroll
  for (int r = 0; r < 8; ++r) {
    red[wave][0][r][lane] = a00[r];
    red[wave][1][r][lane] = a01[r];
    red[wave][2][r][lane] = a10[r];
    red[wave][3][r][lane] = a11[r];
  }
  __syncthreads();

#pragma unroll
  for (int e = 0; e < 4; ++e) {
    const int lin = tid + 256 * e;          // [0, 1024)
    const int lane_ = lin & 31;
    const int r = (lin >> 5) & 7;
    const int t = lin >> 8;                 // tile 0..3
    float s = 0.f;
#pragma unroll
    for (int w = 0; w < 8; ++w) s += red[w][t][r][lane_];
    const int m = (t >> 1) * 16 + r + ((lane_ >= 16) ? 8 : 0);
    const int n = (t & 1) * 16 + (lane_ & 15);
    G[(size_t)b * 1024 + m * 32 + n] = s;
  }
}

// ---------------------------------------------------------------------------
// attn: cosine sim from Gram diag, per-branch topk mask + softmax, alpha-
// weighted combine. One thread per (batch, padded row). att stored 32x32/batch
// with pad rows/cols exactly zero (required by the apply matmul).
// ---------------------------------------------------------------------------
__global__ __launch_bounds__(256) void attn_kernel(
    const float* __restrict__ G, const float* __restrict__ adjn,
    const float* __restrict__ alpha, const float* __restrict__ temps,
    float* __restrict__ attOut, int layer, int topk) {
  const int gid = blockIdx.x * blockDim.x + threadIdx.x;
  if (gid >= NB * 32) return;
  const int b = gid >> 5;
  const int r = gid & 31;
  float* out = attOut + (size_t)b * 1024 + r * 32;
  if (r >= NN) {
    for (int m = 0; m < 32; ++m) out[m] = 0.f;
    return;
  }
  const float* Gb = G + (size_t)b * 1024;
  const float normr = sqrtf(fmaxf(Gb[r * 32 + r], 0.f)) + 1e-6f;

  float sim[NN], accv[NN];
  for (int m = 0; m < NN; ++m) {
    const float normm = sqrtf(fmaxf(Gb[m * 32 + m], 0.f)) + 1e-6f;
    sim[m] = Gb[r * 32 + m] / (normr * normm) * adjn[r * NN + m];
    accv[m] = 0.f;
  }

  for (int k = 0; k < 3; ++k) {
    const float temp = fminf(fmaxf(temps[k], 0.1f), 10.f);
    float av[NN];
    bool sel[NN];
    for (int m = 0; m < NN; ++m) {
      av[m] = sim[m] / temp + (m == r ? 0.1f : 0.f);
      sel[m] = false;
    }
    // top-k selection, first-occurrence tie-break (matches lax.top_k)
    for (int i = 0; i < topk; ++i) {
      float best = -3.4e38f;
      int bi = 0;
      for (int m = 0; m < NN; ++m)
        if (!sel[m] && av[m] > best) {
          best = av[m];
          bi = m;
        }
      sel[bi] = true;
    }
    // softmax over att*mask (masked entries become exactly 0, still in softmax)
    float mk[NN];
    float mx = -3.4e38f;
    for (int m = 0; m < NN; ++m) {
      mk[m] = sel[m] ? av[m] : 0.f;
      mx = fmaxf(mx, mk[m]);
    }
    float e[NN];
    float s = 0.f;
    for (int m = 0; m < NN; ++m) {
      e[m] = expf(mk[m] - mx);
      s += e[m];
    }
    const float a = alpha[layer * 3 + k];
    for (int m = 0; m < NN; ++m) accv[m] += a * e[m] / s;
  }
  for (int m = 0; m < NN; ++m) out[m] = accv[m];
  for (int m = NN; m < 32; ++m) out[m] = 0.f;
}

// ---------------------------------------------------------------------------
// apply: out[b] = att[b](22x22, padded 32x32) @ hin[b](22x4096).
// EPI=0: relu (layer 0).  EPI=1: += x residual (layer 1, runs in-place).
// XL=1: hin read from x with transposed addressing (layer 0).
// grid = (B, 16 column groups of 256); 4 waves/block, 4 col-tiles/wave.
// Only 6 k-steps needed (att K columns >= 24 are all zero). B rows >= 22 are
// clamped: their finite garbage multiplies att's exactly-zero pad columns.
// ---------------------------------------------------------------------------
template <int XL, int EPI>
__global__ __launch_bounds__(128) void apply_kernel(
    const float* __restrict__ hin, const float* __restrict__ att,
    const float* __restrict__ xres, float* __restrict__ out) {
  const int tid = threadIdx.x;
  const int wave = tid >> 5;
  const int lane = tid & 31;
  const int b = blockIdx.x;
  const int l15 = lane & 15;
  const bool hi = lane >= 16;

  const float* attB = att + (size_t)b * 1024;
  v2f afrag[2][6];
#pragma unroll
  for (int mt = 0; mt < 2; ++mt)
#pragma unroll
    for (int ks = 0; ks < 6; ++ks) {
      const int kk = ks * 4 + (hi ? 2 : 0);
      const int row = mt * 16 + l15;
      afrag[mt][ks][0] = attB[row * 32 + kk];
      afrag[mt][ks][1] = attB[row * 32 + kk + 1];
    }

  for (int tile = wave; tile < 16; tile += 4) {
    const int col0 = blockIdx.y * 256 + tile * 16;
    const int c = col0 + l15;
    const int tt = col0 >> 9;      // uniform within the 16-col tile
    const int cc = c & 511;
    v8f acc0 = v8zero(), acc1 = v8zero();

#pragma unroll
    for (int ks = 0; ks < 6; ++ks) {
      const int kk = ks * 4 + (hi ? 2 : 0);       // rows kk, kk+1 of hin
      const int r0 = (kk < NN) ? kk : (NN - 1);   // clamp, no predication
      const int r1 = (kk + 1 < NN) ? (kk + 1) : (NN - 1);
      v2f bf;
      if (XL) {
        bf[0] = hin[(((size_t)b * NT + tt) * NN + r0) * NC + cc];
        bf[1] = hin[(((size_t)b * NT + tt) * NN + r1) * NC + cc];
      } else {
        bf[0] = hin[((size_t)b * NN + r0) * NF + c];
        bf[1] = hin[((size_t)b * NN + r1) * NF + c];
      }
      acc0 = wmma_f32(afrag[0][ks], bf, acc0);
      acc1 = wmma_f32(afrag[1][ks], bf, acc1);
    }

#pragma unroll
    for (int r = 0; r < 8; ++r) {
      const int mlo = r + (hi ? 8 : 0);  // 0..15, always < NN
      {
        float v = acc0[r];
        if (EPI == 0)
          v = fmaxf(v, 0.f);
        else
          v += xres[(((size_t)b * NT + tt) * NN + mlo) * NC + cc];
        out[((size_t)b * NN + mlo) * NF + c] = v;
      }
      const int m1 = 16 + mlo;
      if (m1 < NN) {
        float v = acc1[r];
        if (EPI == 0)
          v = fmaxf(v, 0.f);
        else
          v += xres[(((size_t)b * NT + tt) * NN + m1) * NC + cc];
        out[((size_t)b * NN + m1) * NF + c] = v;
      }
    }
  }
}

// ---------------------------------------------------------------------------
// launch
// ---------------------------------------------------------------------------
extern "C" void kernel_launch(void* const* d_in, const int* in_sizes, int n_in,
                              void* d_out, int out_size, void* d_ws,
                              size_t ws_size, hipStream_t stream) {
  (void)in_sizes;
  (void)n_in;
  (void)out_size;
  (void)ws_size;
  const float* x = (const float*)d_in[0];      // [512, 8, 22, 512]
  const float* adj = (const float*)d_in[1];    // [22, 22]
  const float* temps = (const float*)d_in[2];  // [3]
  const float* flog = (const float*)d_in[3];   // [2, 3]
  float* out = (float*)d_out;                  // [512, 22, 4096]

  float* ws = (float*)d_ws;
  float* ws_adjn = ws;                      // 484 floats
  float* ws_alpha = ws + 512;               // 6 floats
  float* ws_G = ws + 1024;                  // 512*1024 floats (padded 32x32)
  float* ws_att = ws + 1024 + NB * 1024;    // 512*1024 floats (padded 32x32)
  // total workspace: ~4.2 MB

  prep_kernel<<<1, 32, 0, stream>>>(adj, flog, ws_adjn, ws_alpha);

  // layer 0: read x directly (transposed addressing), relu -> d_out
  gram_kernel<1><<<NB, 256, 0, stream>>>(x, ws_G);
  attn_kernel<<<(NB * 32) / 256, 256, 0, stream>>>(ws_G, ws_adjn, ws_alpha,
                                                   temps, ws_att, 0, 10);
  apply_kernel<1, 0><<<dim3(NB, 16), 128, 0, stream>>>(x, ws_att, nullptr, out);

  // layer 1: d_out -> d_out in place (per-column-tile read-before-write),
  // + x residual, no relu
  gram_kernel<0><<<NB, 256, 0, stream>>>(out, ws_G);
  attn_kernel<<<(NB * 32) / 256, 256, 0, stream>>>(ws_G, ws_adjn, ws_alpha,
                                                   temps, ws_att, 1, 3);
  apply_kernel<0, 1><<<dim3(NB, 16), 128, 0, stream>>>(out, ws_att, x, out);
}